// EigenHora_37203006718267
// MI455X (gfx1250) — compile-verified
//
#include <hip/hip_runtime.h>

// out[s,f] = sum_l sum_h sum_w jac_l[s,h,w] * ul_l[h,f] * ur_l[w,f]
// S=32, D=1, F=16. Layers (H,W): (2048,1024), (2048,2048), (1024,2048).
//
// WMMA mapping (V_WMMA_F32_16X16X4_F32, D = A(16x4) * B(4x16) + C):
//   M = f (16 features), K = w (4 per instr), N = h_local (16 rows of a tile)
//   A[f,k]  = ur[w+k, f]          (hot in cache, 2x b32 per lane)
//   B[k,n]  = jac[s, h0+n, w+k]   (streamed, 1x non-temporal b64 per lane)
//   C[f,n] += sum_k A*B           -> after W loop: C[f,n] = sum_w ur[w,f]*jac[s,h0+n,w]
// Epilogue folds C with ul and atomically accumulates into out[32x16].

typedef float v2f __attribute__((ext_vector_type(2)));
typedef float v8f __attribute__((ext_vector_type(8)));

#define WAVES_L1 4096   // 32 samples * (2048/16) h-tiles
#define WAVES_L2 4096   // 32 * 128
#define WAVES_L3 2048   // 32 * (1024/16)
#define TOTAL_WAVES (WAVES_L1 + WAVES_L2 + WAVES_L3)   // 10240
#define BLOCKS (TOTAL_WAVES / 8)                       // 1280 blocks of 8 waves

__global__ void eigenhora_zero(float* __restrict__ out, int n) {
    int i = blockIdx.x * blockDim.x + threadIdx.x;
    if (i < n) out[i] = 0.0f;
}

__global__ __launch_bounds__(256) void eigenhora_wmma(
    const float* __restrict__ jac1, const float* __restrict__ ul1, const float* __restrict__ ur1,
    const float* __restrict__ jac2, const float* __restrict__ ul2, const float* __restrict__ ur2,
    const float* __restrict__ jac3, const float* __restrict__ ul3, const float* __restrict__ ur3,
    float* __restrict__ out)
{
    const int lane  = threadIdx.x & 31;
    const int gwave = blockIdx.x * 8 + (threadIdx.x >> 5);

    const float* jac; const float* ul; const float* ur;
    int H, W, id;
    if (gwave < WAVES_L1) {
        jac = jac1; ul = ul1; ur = ur1; H = 2048; W = 1024; id = gwave;
    } else if (gwave < WAVES_L1 + WAVES_L2) {
        jac = jac2; ul = ul2; ur = ur2; H = 2048; W = 2048; id = gwave - WAVES_L1;
    } else {
        jac = jac3; ul = ul3; ur = ur3; H = 1024; W = 2048; id = gwave - (WAVES_L1 + WAVES_L2);
    }

    const int nHT = H >> 4;          // h-tiles per sample
    const int s   = id / nHT;        // sample
    const int h0  = (id % nHT) << 4; // first h row of this wave's tile

    const int n     = lane & 15;     // B: h_local / A: f (m index)
    const int khalf = lane >> 4;     // selects K-pair {0,1} or {2,3}

    // B: jac[s, h0+n, w + 2*khalf + {0,1}]  -> one b64 per wmma, 8B aligned
    const float* brow = jac + ((size_t)s * H + (size_t)(h0 + n)) * (size_t)W + 2 * khalf;
    // A: ur[(w + 2*khalf + {0,1})*16 + f]
    const float* aptr = ur + 32 * khalf + n;

    v8f c0 = {};
    v8f c1 = {};

    for (int w = 0; w < W; w += 8) {
        v2f b0 = __builtin_nontemporal_load((const v2f*)(brow + w));
        v2f b1 = __builtin_nontemporal_load((const v2f*)(brow + w + 4));
        v2f a0, a1;
        a0.x = aptr[w * 16];
        a0.y = aptr[w * 16 + 16];
        a1.x = aptr[(w + 4) * 16];
        a1.y = aptr[(w + 4) * 16 + 16];
        // two independent accumulator chains to cover wmma->wmma RAW latency
        c0 = __builtin_amdgcn_wmma_f32_16x16x4_f32(false, a0, false, b0,
                                                   (short)0, c0, false, false);
        c1 = __builtin_amdgcn_wmma_f32_16x16x4_f32(false, a1, false, b1,
                                                   (short)0, c1, false, false);
    }

    // C layout: lane L, VGPR j holds C[f = j + 8*(L>>4), h_local = L&15]
    const float* ulrow = ul + (size_t)(h0 + n) * 16 + 8 * khalf;
    float p[8];
#pragma unroll
    for (int j = 0; j < 8; ++j)
        p[j] = (c0[j] + c1[j]) * ulrow[j];

    // reduce over the 16 lanes of each half (masks < 16 stay inside a half)
#pragma unroll
    for (int m = 1; m < 16; m <<= 1) {
#pragma unroll
        for (int j = 0; j < 8; ++j)
            p[j] += __shfl_xor(p[j], m, 32);
    }

    if (n == 0) {
        float* o = out + s * 16 + 8 * khalf;  // lane 0 -> f 0..7, lane 16 -> f 8..15
#pragma unroll
        for (int j = 0; j < 8; ++j)
            unsafeAtomicAdd(o + j, p[j]);     // global_atomic_add_f32
    }
}

extern "C" void kernel_launch(void* const* d_in, const int* in_sizes, int n_in,
                              void* d_out, int out_size, void* d_ws, size_t ws_size,
                              hipStream_t stream) {
    float* out = (float*)d_out;
    // d_out is poisoned and atomically accumulated into: zero it first.
    eigenhora_zero<<<(out_size + 255) / 256, 256, 0, stream>>>(out, out_size);
    eigenhora_wmma<<<BLOCKS, 256, 0, stream>>>(
        (const float*)d_in[0], (const float*)d_in[1], (const float*)d_in[2],
        (const float*)d_in[3], (const float*)d_in[4], (const float*)d_in[5],
        (const float*)d_in[6], (const float*)d_in[7], (const float*)d_in[8],
        out);
}